// Finsentry_69286412419323
// MI455X (gfx1250) — compile-verified
//
#include <hip/hip_runtime.h>
#include <hip/hip_bf16.h>
#include <float.h>

#define HD 64

typedef float v2f __attribute__((ext_vector_type(2)));
typedef float v8f __attribute__((ext_vector_type(8)));

__device__ __forceinline__ float lrelu(float x) { return x > 0.f ? x : 0.2f * x; }

// monotonic float -> uint key (orderable), for atomicMax-based segment max
__device__ __forceinline__ unsigned fkey(float x) {
    unsigned u = __float_as_uint(x);
    return (u & 0x80000000u) ? ~u : (u | 0x80000000u);
}
__device__ __forceinline__ float funkey(unsigned k) {
    unsigned u = (k & 0x80000000u) ? (k & 0x7fffffffu) : ~k;
    return __uint_as_float(u);
}

// ---------------------------------------------------------------------------
// Dense: H[m][n] = sum_k X[m][k] * W[k][n];  X: (Mtiles*16) x 64, W: 64 x 64.
// One wave per 16x16 output tile, fp32 WMMA, K stepped by 4 (16 wmma ops/tile).
// ---------------------------------------------------------------------------
__global__ void k_wmma_lin64(const float* __restrict__ X, const float* __restrict__ W,
                             float* __restrict__ Hout, int Mtiles) {
    int wave  = (blockIdx.x * blockDim.x + threadIdx.x) >> 5;
    int lane  = threadIdx.x & 31;
    int ntile = wave & 3;          // 64 cols = 4 tiles
    int mtile = wave >> 2;
    if (mtile >= Mtiles) return;   // wave-uniform exit, EXEC stays all-ones inside

    int mrow  = (lane & 15);
    int col   = ntile * 16 + (lane & 15);
    int khalf = lane >> 4;         // 0: k={0,1}, 1: k={2,3} within each K4 step

    v8f acc = {};
    const float* xptr = X + (size_t)(mtile * 16 + mrow) * HD + 2 * khalf;
    const float* wptr = W + (size_t)(2 * khalf) * HD + col;
#pragma unroll
    for (int ks = 0; ks < 16; ++ks) {
        v2f a = *(const v2f*)xptr;               // A[m][4ks+2*khalf + {0,1}]
        v2f b;
        b.x = wptr[0];                           // W[4ks+2*khalf+0][col]
        b.y = wptr[HD];                          // W[4ks+2*khalf+1][col]
        acc = __builtin_amdgcn_wmma_f32_16x16x4_f32(
            /*neg_a=*/false, a, /*neg_b=*/false, b,
            /*c_mod=*/(short)0, acc, /*reuse_a=*/false, /*reuse_b=*/false);
        xptr += 4;
        wptr += 4 * HD;
    }
    // D layout: VGPR j -> row (j + 8*khalf), col = lane&15
    int r0 = mtile * 16 + 8 * khalf;
#pragma unroll
    for (int j = 0; j < 8; ++j)
        Hout[(size_t)(r0 + j) * HD + col] = acc[j];
}

// ---------------------------------------------------------------------------
// Degrees / norms
// ---------------------------------------------------------------------------
__global__ void k_fill_ones2(float* a, float* b, int n) {
    int i = blockIdx.x * blockDim.x + threadIdx.x;
    if (i < n) { a[i] = 1.0f; b[i] = 1.0f; }   // self-loop contributes 1 to both degrees
}
__global__ void k_deg_count(const int* __restrict__ src, const int* __restrict__ dst,
                            float* dego, float* degi, int e_cnt) {
    int i = blockIdx.x * blockDim.x + threadIdx.x;
    if (i >= e_cnt) return;
    atomicAdd(&dego[src[i]], 1.0f);
    atomicAdd(&degi[dst[i]], 1.0f);
}
__global__ void k_rsqrt2(float* a, float* b, int n) {
    int i = blockIdx.x * blockDim.x + threadIdx.x;
    if (i < n) { a[i] = rsqrtf(a[i]); b[i] = rsqrtf(b[i]); }
}

// ---------------------------------------------------------------------------
// GCN aggregation
// ---------------------------------------------------------------------------
__global__ void k_copy4(const float* __restrict__ in, float* __restrict__ out, long long n4) {
    long long i = (long long)blockIdx.x * blockDim.x + threadIdx.x;
    if (i < n4) ((float4*)out)[i] = ((const float4*)in)[i];
}
// agg[i] = h[i]*no[i]   (self-loop term, also zero-free init)
__global__ void k_gcn_selfinit(const float* __restrict__ h, const float* __restrict__ no,
                               float* __restrict__ agg, int n) {
    int gid = blockIdx.x * blockDim.x + threadIdx.x;
    int i = gid >> 4, q = gid & 15;
    if (i >= n) return;
    float s = no[i];
    float4 v = *(const float4*)(h + (size_t)i * HD + 4 * q);
    v.x *= s; v.y *= s; v.z *= s; v.w *= s;
    *(float4*)(agg + (size_t)i * HD + 4 * q) = v;
}
// one wave per edge, 2 features per lane: agg[dst] += h[src]*no[src]
__global__ void k_gcn_edges(const float* __restrict__ h, const float* __restrict__ no,
                            const int* __restrict__ src, const int* __restrict__ dst,
                            float* __restrict__ agg, int e_cnt) {
    long long gid = (long long)blockIdx.x * blockDim.x + threadIdx.x;
    int e = (int)(gid >> 5), lane = (int)(gid & 31);
    if (e >= e_cnt) return;
    int s = src[e], d = dst[e];
    float ns = no[s];
    v2f hv = *(const v2f*)(h + (size_t)s * HD + 2 * lane);
    float* o = agg + (size_t)d * HD + 2 * lane;
    atomicAdd(o, hv.x * ns);
    atomicAdd(o + 1, hv.y * ns);
}
// x[i] = relu(agg[i]*ni[i] + b)
__global__ void k_gcn_final(const float* __restrict__ agg, const float* __restrict__ ni,
                            const float* __restrict__ b, float* __restrict__ x, int n) {
    int gid = blockIdx.x * blockDim.x + threadIdx.x;
    int i = gid >> 4, q = gid & 15;
    if (i >= n) return;
    float s = ni[i];
    float4 a = *(const float4*)(agg + (size_t)i * HD + 4 * q);
    float4 bb = *(const float4*)(b + 4 * q);
    float4 o;
    o.x = fmaxf(a.x * s + bb.x, 0.f);
    o.y = fmaxf(a.y * s + bb.y, 0.f);
    o.z = fmaxf(a.z * s + bb.z, 0.f);
    o.w = fmaxf(a.w * s + bb.w, 0.f);
    *(float4*)(x + (size_t)i * HD + 4 * q) = o;
}

// ---------------------------------------------------------------------------
// Readout: per-feature mean & max over n nodes. grid = 64 blocks (one/feature)
// ---------------------------------------------------------------------------
__global__ void k_readout(const float* __restrict__ x, float* __restrict__ ro, int n) {
    int f = blockIdx.x;
    float sum = 0.f, mx = -FLT_MAX;
    for (int i = threadIdx.x; i < n; i += blockDim.x) {
        float v = x[(size_t)i * HD + f];
        sum += v;
        mx = fmaxf(mx, v);
    }
    __shared__ float ssum[256], smax[256];
    ssum[threadIdx.x] = sum; smax[threadIdx.x] = mx;
    __syncthreads();
    for (int s = 128; s > 0; s >>= 1) {
        if (threadIdx.x < s) {
            ssum[threadIdx.x] += ssum[threadIdx.x + s];
            smax[threadIdx.x] = fmaxf(smax[threadIdx.x], smax[threadIdx.x + s]);
        }
        __syncthreads();
    }
    if (threadIdx.x == 0) {
        ro[f]      = ssum[0] / (float)n;   // mean
        ro[HD + f] = smax[0];              // max
    }
}

// ---------------------------------------------------------------------------
// small dense: out[j] = (relu?) in(1xK) @ W(KxNcol) + b[j]
// ---------------------------------------------------------------------------
__global__ void k_mm1xk(const float* __restrict__ in, const float* __restrict__ W,
                        const float* __restrict__ b, float* __restrict__ out,
                        int K, int Ncol, int do_relu) {
    int j = blockIdx.x * blockDim.x + threadIdx.x;
    if (j >= Ncol) return;
    float acc = b ? b[j] : 0.f;
    for (int k = 0; k < K; ++k) acc += in[k] * W[(size_t)k * Ncol + j];
    out[j] = do_relu ? fmaxf(acc, 0.f) : acc;
}

// ---------------------------------------------------------------------------
// GAT pieces (nodes 0..n-1, supernode row n in h/el)
// ---------------------------------------------------------------------------
// el/er dots, one wave per node (wave32), 2 features per lane
__global__ void k_eler(const float* __restrict__ h, const float* __restrict__ al,
                       const float* __restrict__ ar, float* __restrict__ el,
                       float* __restrict__ er, int n_tot) {
    long long gid = (long long)blockIdx.x * blockDim.x + threadIdx.x;
    int i = (int)(gid >> 5), lane = (int)(gid & 31);
    if (i >= n_tot) return;
    v2f hv = *(const v2f*)(h + (size_t)i * HD + 2 * lane);
    v2f av = *(const v2f*)(al + 2 * lane);
    v2f rv = *(const v2f*)(ar + 2 * lane);
    float pl = hv.x * av.x + hv.y * av.y;
    float pr = hv.x * rv.x + hv.y * rv.y;
    for (int off = 16; off > 0; off >>= 1) {
        pl += __shfl_down(pl, off, 32);
        pr += __shfl_down(pr, off, 32);
    }
    if (lane == 0) { el[i] = pl; er[i] = pr; }
}
// init max-key with self-loop edge and supernode edge
__global__ void k_gat_minit(const float* __restrict__ el, const float* __restrict__ er,
                            unsigned* __restrict__ mkey, int n) {
    int i = blockIdx.x * blockDim.x + threadIdx.x;
    if (i >= n) return;
    float eri = er[i];
    float es = lrelu(el[i] + eri);
    float ep = lrelu(el[n] + eri);   // supernode src = n
    mkey[i] = fkey(fmaxf(es, ep));
}
__global__ void k_gat_medges(const float* __restrict__ el, const float* __restrict__ er,
                             const int* __restrict__ src, const int* __restrict__ dst,
                             unsigned* __restrict__ mkey, int e_cnt) {
    int i = blockIdx.x * blockDim.x + threadIdx.x;
    if (i >= e_cnt) return;
    int s = src[i], d = dst[i];
    atomicMax(&mkey[d], fkey(lrelu(el[s] + er[d])));
}
// init accumulators with self-loop + supernode contributions; convert mkey->float
__global__ void k_gat_accinit(const float* __restrict__ h, const float* __restrict__ el,
                              const float* __restrict__ er, const unsigned* __restrict__ mkey,
                              float* __restrict__ mflt, float* __restrict__ ssum,
                              float* __restrict__ acc, int n) {
    long long gid = (long long)blockIdx.x * blockDim.x + threadIdx.x;
    int i = (int)(gid >> 5), lane = (int)(gid & 31);
    if (i >= n) return;
    float m = funkey(mkey[i]);
    float eri = er[i];
    float ees = __expf(lrelu(el[i] + eri) - m);
    float eep = __expf(lrelu(el[n] + eri) - m);
    if (lane == 0) { mflt[i] = m; ssum[i] = ees + eep; }
    v2f hi = *(const v2f*)(h + (size_t)i * HD + 2 * lane);
    v2f hs = *(const v2f*)(h + (size_t)n * HD + 2 * lane);
    v2f o;
    o.x = ees * hi.x + eep * hs.x;
    o.y = ees * hi.y + eep * hs.y;
    *(v2f*)(acc + (size_t)i * HD + 2 * lane) = o;
}
// one wave per edge: ee = exp(e - m[dst]); ssum[dst]+=ee; acc[dst]+=ee*h[src]
__global__ void k_gat_accedges(const float* __restrict__ h, const float* __restrict__ el,
                               const float* __restrict__ er, const float* __restrict__ mflt,
                               const int* __restrict__ src, const int* __restrict__ dst,
                               float* __restrict__ ssum, float* __restrict__ acc, int e_cnt) {
    long long gid = (long long)blockIdx.x * blockDim.x + threadIdx.x;
    int e = (int)(gid >> 5), lane = (int)(gid & 31);
    if (e >= e_cnt) return;
    int s = src[e], d = dst[e];
    float ee = __expf(lrelu(el[s] + er[d]) - mflt[d]);
    if (lane == 0) atomicAdd(&ssum[d], ee);
    v2f hv = *(const v2f*)(h + (size_t)s * HD + 2 * lane);
    float* o = acc + (size_t)d * HD + 2 * lane;
    atomicAdd(o, ee * hv.x);
    atomicAdd(o + 1, ee * hv.y);
}
// x[i] = acc[i]/ssum[i] + b   (no relu)
__global__ void k_gat_final(const float* __restrict__ acc, const float* __restrict__ ssum,
                            const float* __restrict__ b, float* __restrict__ x, int n) {
    int gid = blockIdx.x * blockDim.x + threadIdx.x;
    int i = gid >> 4, q = gid & 15;
    if (i >= n) return;
    float inv = 1.0f / ssum[i];
    float4 a = *(const float4*)(acc + (size_t)i * HD + 4 * q);
    float4 bb = *(const float4*)(b + 4 * q);
    float4 o;
    o.x = a.x * inv + bb.x;
    o.y = a.y * inv + bb.y;
    o.z = a.z * inv + bb.z;
    o.w = a.w * inv + bb.w;
    *(float4*)(x + (size_t)i * HD + 4 * q) = o;
}

// final: logits (1x96 @ 96x2) + log_softmax -> d_out[0..1]
__global__ void k_final_logsoftmax(const float* __restrict__ in, const float* __restrict__ W3,
                                   const float* __restrict__ b3, float* __restrict__ out) {
    __shared__ float lg[2];
    int t = threadIdx.x;
    if (t < 2) {
        float acc = b3[t];
        for (int k = 0; k < 96; ++k) acc += in[k] * W3[k * 2 + t];
        lg[t] = acc;
    }
    __syncthreads();
    if (t == 0) {
        float m = fmaxf(lg[0], lg[1]);
        float lz = m + __logf(__expf(lg[0] - m) + __expf(lg[1] - m));
        out[0] = lg[0] - lz;
        out[1] = lg[1] - lz;
    }
}

// ---------------------------------------------------------------------------
extern "C" void kernel_launch(void* const* d_in, const int* in_sizes, int n_in,
                              void* d_out, int out_size, void* d_ws, size_t ws_size,
                              hipStream_t stream) {
    const int N = in_sizes[0] / HD;     // 50000
    const int E = in_sizes[21];         // 800000
    const int NL = 3;
    const int Mtiles = N / 16;

    const float* x_in[3] = { (const float*)d_in[0], (const float*)d_in[1], (const float*)d_in[2] };
    const float* Wc[3]   = { (const float*)d_in[3], (const float*)d_in[5], (const float*)d_in[7] };
    const float* bc[3]   = { (const float*)d_in[4], (const float*)d_in[6], (const float*)d_in[8] };
    const float* Wx      = (const float*)d_in[9];
    const float* bx      = (const float*)d_in[10];
    const float* Wgat    = (const float*)d_in[11];
    const float* al      = (const float*)d_in[12];
    const float* ar      = (const float*)d_in[13];
    const float* bgat    = (const float*)d_in[14];
    const float* W1      = (const float*)d_in[15];
    const float* b1      = (const float*)d_in[16];
    const float* W2      = (const float*)d_in[17];
    const float* b2      = (const float*)d_in[18];
    const float* W3      = (const float*)d_in[19];
    const float* b3      = (const float*)d_in[20];
    const int* srcs[3]   = { (const int*)d_in[21], (const int*)d_in[23], (const int*)d_in[25] };
    const int* dsts[3]   = { (const int*)d_in[22], (const int*)d_in[24], (const int*)d_in[26] };
    float* out = (float*)d_out;

    // workspace layout (floats), all offsets multiples of 16 floats
    float* ws = (float*)d_ws;
    const size_t NH = (size_t)N * HD;
    size_t off = 0;
    float* xb[3];
    xb[0] = ws + off; off += NH;
    xb[1] = ws + off; off += NH;
    xb[2] = ws + off; off += NH;
    float* hbuf = ws + off; off += NH + HD;           // (N+1) x 64
    float* agg  = ws + off; off += NH;
    float* nob[3]; float* nib[3];
    for (int g = 0; g < 3; ++g) {
        nob[g] = ws + off; off += (size_t)((N + 15) & ~15);
        nib[g] = ws + off; off += (size_t)((N + 15) & ~15);
    }
    float* el   = ws + off; off += (size_t)((N + 16) & ~15);
    float* er   = ws + off; off += (size_t)((N + 16) & ~15);
    unsigned* mkey = (unsigned*)(ws + off); off += (size_t)((N + 15) & ~15);
    float* mflt = ws + off; off += (size_t)((N + 15) & ~15);
    float* ssum = ws + off; off += (size_t)((N + 15) & ~15);
    float* robase = ws + off; off += 3 * 2 * HD;      // [ro_s | ro_g | ro_t] = 384 contiguous
    float* rob[3] = { robase, robase + 2 * HD, robase + 4 * HD };
    float* fb[3];
    fb[0] = ws + off; off += HD;
    fb[1] = ws + off; off += HD;
    fb[2] = ws + off; off += HD;
    float* m1 = ws + off; off += 192;
    float* m2 = ws + off; off += 96;

    const int BT = 256;
    auto blk  = [&](long long t) { return dim3((unsigned)((t + BT - 1) / BT)); };
    const long long nodeF4   = (long long)N * 16;       // float4-granular node work
    const long long edgeWave = (long long)E * 32;       // wave-per-edge work
    const long long nodeWave = (long long)N * 32;
    const long long nodeWaveS = (long long)(N + 1) * 32;
    const dim3 wmmaGrid((unsigned)((Mtiles * 4 + 7) / 8));   // 8 waves (256 thr) per block

    // --- degree norms (per graph) ---
    for (int g = 0; g < 3; ++g) {
        k_fill_ones2<<<blk(N), BT, 0, stream>>>(nob[g], nib[g], N);
        k_deg_count<<<blk(E), BT, 0, stream>>>(srcs[g], dsts[g], nob[g], nib[g], E);
        k_rsqrt2<<<blk(N), BT, 0, stream>>>(nob[g], nib[g], N);
    }
    // --- copy initial node features into working buffers ---
    for (int g = 0; g < 3; ++g)
        k_copy4<<<blk(NH / 4), BT, 0, stream>>>(x_in[g], xb[g], (long long)(NH / 4));

    // --- main layers ---
    for (int it = 0; it < NL - 1; ++it) {
        // ConvPool per graph: x <- relu(GCN(x)), readout
        for (int g = 0; g < 3; ++g) {
            k_wmma_lin64<<<wmmaGrid, BT, 0, stream>>>(xb[g], Wc[g] + (size_t)it * HD * HD, hbuf, Mtiles);
            k_gcn_selfinit<<<blk(nodeF4), BT, 0, stream>>>(hbuf, nob[g], agg, N);
            k_gcn_edges<<<blk(edgeWave), BT, 0, stream>>>(hbuf, nob[g], srcs[g], dsts[g], agg, E);
            k_gcn_final<<<blk(nodeF4), BT, 0, stream>>>(agg, nib[g], bc[g] + (size_t)it * HD, xb[g], N);
            k_readout<<<HD, BT, 0, stream>>>(xb[g], rob[g], N);
        }
        // cross features (1x128 @ 128x64 + relu)
        if ((it & 1) == 0) {
            k_mm1xk<<<1, BT, 0, stream>>>(rob[2], Wx + 0 * 128 * HD, bx + 0 * HD, fb[1], 128, HD, 1); // t2g
            k_mm1xk<<<1, BT, 0, stream>>>(rob[1], Wx + 1 * 128 * HD, bx + 1 * HD, fb[0], 128, HD, 1); // g2s
            k_mm1xk<<<1, BT, 0, stream>>>(rob[0], Wx + 2 * 128 * HD, bx + 2 * HD, fb[2], 128, HD, 1); // s2t
        } else {
            k_mm1xk<<<1, BT, 0, stream>>>(rob[0], Wx + 3 * 128 * HD, bx + 3 * HD, fb[1], 128, HD, 1); // s2g
            k_mm1xk<<<1, BT, 0, stream>>>(rob[1], Wx + 4 * 128 * HD, bx + 4 * HD, fb[2], 128, HD, 1); // g2t
            k_mm1xk<<<1, BT, 0, stream>>>(rob[2], Wx + 5 * 128 * HD, bx + 5 * HD, fb[0], 128, HD, 1); // t2s
        }
        // GAT with supernode per graph
        for (int g = 0; g < 3; ++g) {
            const float* Wg  = Wgat + (size_t)g * HD * HD;
            const float* alg = al + (size_t)g * HD;
            const float* arg = ar + (size_t)g * HD;
            const float* bg  = bgat + (size_t)g * HD;
            k_wmma_lin64<<<wmmaGrid, BT, 0, stream>>>(xb[g], Wg, hbuf, Mtiles);
            // supernode row: h[N] = f @ W
            k_mm1xk<<<1, BT, 0, stream>>>(fb[g], Wg, nullptr, hbuf + (size_t)N * HD, HD, HD, 0);
            k_eler<<<blk(nodeWaveS), BT, 0, stream>>>(hbuf, alg, arg, el, er, N + 1);
            k_gat_minit<<<blk(N), BT, 0, stream>>>(el, er, mkey, N);
            k_gat_medges<<<blk(E), BT, 0, stream>>>(el, er, srcs[g], dsts[g], mkey, E);
            k_gat_accinit<<<blk(nodeWave), BT, 0, stream>>>(hbuf, el, er, mkey, mflt, ssum, agg, N);
            k_gat_accedges<<<blk(edgeWave), BT, 0, stream>>>(hbuf, el, er, mflt, srcs[g], dsts[g], ssum, agg, E);
            k_gat_final<<<blk(nodeF4), BT, 0, stream>>>(agg, ssum, bg, xb[g], N);
        }
    }

    // --- final ConvPool (readouts only matter) ---
    for (int g = 0; g < 3; ++g) {
        k_wmma_lin64<<<wmmaGrid, BT, 0, stream>>>(xb[g], Wc[g] + (size_t)(NL - 1) * HD * HD, hbuf, Mtiles);
        k_gcn_selfinit<<<blk(nodeF4), BT, 0, stream>>>(hbuf, nob[g], agg, N);
        k_gcn_edges<<<blk(edgeWave), BT, 0, stream>>>(hbuf, nob[g], srcs[g], dsts[g], agg, E);
        k_gcn_final<<<blk(nodeF4), BT, 0, stream>>>(agg, nib[g], bc[g] + (size_t)(NL - 1) * HD, xb[g], N);
        k_readout<<<HD, BT, 0, stream>>>(xb[g], rob[g], N);
    }

    // --- classifier MLP: 384 -> 192 -> 96 -> 2 + log_softmax ---
    k_mm1xk<<<1, BT, 0, stream>>>(robase, W1, b1, m1, 384, 192, 1);
    k_mm1xk<<<1, BT, 0, stream>>>(m1, W2, b2, m2, 192, 96, 1);
    k_final_logsoftmax<<<1, 64, 0, stream>>>(m2, W3, b3, out);
    (void)n_in; (void)out_size; (void)ws_size; (void)off;
}